// GCN_32366873542801
// MI455X (gfx1250) — compile-verified
//
#include <hip/hip_runtime.h>
#include <hip/hip_bf16.h>

// ---------------------------------------------------------------------------
// GCN (2x GCNConv + global_add_pool + linear head) for gfx1250 (MI455X).
//  * GEMMs: V_WMMA_F32_16X16X4_F32 (native f32 WMMA, wave32), W staged into
//    LDS once per block via the Tensor Data Mover (tensor_load_to_lds +
//    s_wait_tensorcnt), B fragments served from LDS (ds_load).
//  * Edge aggregation: float4 gathers + global_atomic_add_f32 (L2-resident).
// ---------------------------------------------------------------------------

typedef __attribute__((ext_vector_type(2))) float v2f;
typedef __attribute__((ext_vector_type(8))) float v8f;
typedef __attribute__((ext_vector_type(4))) unsigned int u32x4;
typedef __attribute__((ext_vector_type(8))) int i32x8;
typedef __attribute__((ext_vector_type(4))) int i32x4;

#define FEAT 64

__device__ __forceinline__ void atomicAddF(float* p, float v) {
  __hip_atomic_fetch_add(p, v, __ATOMIC_RELAXED, __HIP_MEMORY_SCOPE_AGENT);
}

// ---------------- degree / normalization ----------------------------------

__global__ void gcn_init_deg(unsigned* __restrict__ deg, int n) {
  int i = blockIdx.x * blockDim.x + threadIdx.x;
  if (i < n) deg[i] = 1u;  // self-loop
}

__global__ void gcn_count_deg(const int* __restrict__ ei, unsigned* __restrict__ deg, int E) {
  int e = blockIdx.x * blockDim.x + threadIdx.x;
  if (e < E) atomicAdd(&deg[ei[E + e]], 1u);  // col = target
}

__global__ void gcn_dinv(const unsigned* __restrict__ deg, float* __restrict__ dinv, int n) {
  int i = blockIdx.x * blockDim.x + threadIdx.x;
  if (i < n) dinv[i] = rsqrtf((float)deg[i]);  // deg >= 1 always
}

// ---------------- TDM: stage W[64x64] f32 into LDS -------------------------
// 2-D descriptor: data_size=4B (code 2), tile = tensor = 64x64,
// tensor_dim0_stride = 64 elements. Groups 2/3 zero (<=2D tensor).
// Group0: [1:0]=count=1 | [63:32]=lds_addr | [120:64]=global_addr |
//         [127:126]=type=2.
// Toolchain note: this clang-23 lane's builtin takes 6 args
// (g0, g1, g2, g3, extra i32x8, cpol).

__device__ __forceinline__ void tdm_load_w64x64(const float* W, void* ldsPtr) {
  const unsigned long long ga = (unsigned long long)(uintptr_t)W;
  const unsigned lds = (unsigned)(uintptr_t)ldsPtr;  // flat addr[31:0] == LDS byte offset

  u32x4 g0;
  g0[0] = 1u;                                           // count=1, user mode
  g0[1] = lds;                                          // lds_addr
  g0[2] = (unsigned)(ga & 0xFFFFFFFFu);                 // global_addr[31:0]
  g0[3] = (unsigned)((ga >> 32) & 0x01FFFFFFu)          // global_addr[56:32]
        | (2u << 30);                                   // type = 2 ("image")

  i32x8 g1;
  g1[0] = (int)(2u << 16);        // workgroup_mask=0, data_size=2 (4 bytes)
  g1[1] = (int)(64u << 16);       // tensor_dim0[15:0]=64 (bits 63:48)
  g1[2] = (int)(64u << 16);       // tensor_dim0 hi=0 | tensor_dim1[15:0]=64 (95:80)
  g1[3] = (int)(64u << 16);       // tensor_dim1 hi=0 | tile_dim0=64 (127:112)
  g1[4] = (int)64u;               // tile_dim1=64 (143:128), tile_dim2=0
  g1[5] = (int)64u;               // tensor_dim0_stride[31:0]=64
  g1[6] = (int)(4096u << 16);     // dim0_stride[47:32]=0 | dim1_stride[15:0]=4096
  g1[7] = 0;                      // dim1_stride[47:16]=0

  i32x4 g2 = {0, 0, 0, 0};
  i32x4 g3 = {0, 0, 0, 0};
  i32x8 g4 = {0, 0, 0, 0, 0, 0, 0, 0};

  __builtin_amdgcn_tensor_load_to_lds(g0, g1, g2, g3, g4, 0);
}

// ---------------- WMMA GEMM: OUT[M,64] = A[M,64] @ W[64,64] ----------------
// One wave32 computes a 16-row x 64-col tile; 256-thread block = 8 waves =
// 128 rows. K=64 fully unrolled: 16 steps x 4 col-tiles of
// v_wmma_f32_16x16x4_f32. W served from LDS (TDM-staged).
//
// A (16x4 f32) VGPR layout (ISA 7.12.2): lanes 0-15 hold K={k0,k0+1},
// lanes 16-31 hold K={k0+2,k0+3}, row M = lane&15.
// B (4x16) mirrors: VGPR j, lane-half h -> B[k0 + j + 2*h][N = lane&15].
// C/D (16x16 f32): VGPR r -> lanes 0-15 (M=r, N=lane), lanes 16-31 (M=r+8).

__global__ void gcn_gemm64_wmma(const float* __restrict__ A,
                                const float* __restrict__ W,
                                float* __restrict__ OUT, int M) {
  __shared__ float lw[FEAT * FEAT];  // 16 KB staged weights

  // Wave 0 issues the TDM load and waits its TENSORcnt; barrier publishes.
  if (threadIdx.x < 32) {
    tdm_load_w64x64(W, (void*)lw);
    __builtin_amdgcn_s_wait_tensorcnt(0);
  }
  __syncthreads();

  const int lane = threadIdx.x & 31;
  const int wave = threadIdx.x >> 5;
  const int rowBase = blockIdx.x * 128 + wave * 16;
  const int mr = lane & 15;
  const int kh = lane >> 4;  // 0 or 1: which K pair this half-wave holds

  int arow = rowBase + mr;
  if (arow > M - 1) arow = M - 1;  // clamp loads for the tail; stores guarded
  const float* Arow = A + (size_t)arow * FEAT;

  v8f acc0 = {}, acc1 = {}, acc2 = {}, acc3 = {};

#pragma unroll
  for (int k0 = 0; k0 < FEAT; k0 += 4) {
    const int ka = k0 + kh * 2;

    v2f a;
    {
      const float2 t = *(const float2*)(Arow + ka);
      a.x = t.x; a.y = t.y;
    }

    const float* Wk0 = lw + ka * FEAT;  // row ka  (LDS)
    const float* Wk1 = Wk0 + FEAT;      // row ka+1
    v2f b0, b1, b2, b3;
    b0.x = Wk0[mr +  0]; b0.y = Wk1[mr +  0];
    b1.x = Wk0[mr + 16]; b1.y = Wk1[mr + 16];
    b2.x = Wk0[mr + 32]; b2.y = Wk1[mr + 32];
    b3.x = Wk0[mr + 48]; b3.y = Wk1[mr + 48];

    acc0 = __builtin_amdgcn_wmma_f32_16x16x4_f32(false, a, false, b0, (short)0, acc0, false, false);
    acc1 = __builtin_amdgcn_wmma_f32_16x16x4_f32(false, a, false, b1, (short)0, acc1, false, false);
    acc2 = __builtin_amdgcn_wmma_f32_16x16x4_f32(false, a, false, b2, (short)0, acc2, false, false);
    acc3 = __builtin_amdgcn_wmma_f32_16x16x4_f32(false, a, false, b3, (short)0, acc3, false, false);
  }

#pragma unroll
  for (int r = 0; r < 8; ++r) {
    const int orow = rowBase + r + kh * 8;
    if (orow < M) {
      float* o = OUT + (size_t)orow * FEAT + mr;
      o[0]  = acc0[r];
      o[16] = acc1[r];
      o[32] = acc2[r];
      o[48] = acc3[r];
    }
  }
}

// ---------------- zero fill ------------------------------------------------

__global__ void gcn_zero_f32(float* __restrict__ p, int n) {
  int i = blockIdx.x * blockDim.x + threadIdx.x;
  if (i < n) p[i] = 0.0f;
}

// ---------------- edge scatter: AGG[col] += norm * H[row] ------------------
// 16 lanes per edge, float4 per lane (global_load_b128 gather + 4 f32 atomics).

__global__ void gcn_scatter(const int* __restrict__ ei,
                            const float* __restrict__ dinv,
                            const float* __restrict__ H,
                            float* __restrict__ AGG, int E) {
  const int t = blockIdx.x * blockDim.x + threadIdx.x;
  const int e = t >> 4;
  const int lane = t & 15;
  if (e >= E) return;
  const int r = ei[e];
  const int c = ei[E + e];
  const float nrm = dinv[r] * dinv[c];
  const float4 hv = *(const float4*)(H + (size_t)r * FEAT + lane * 4);
  float* dst = AGG + (size_t)c * FEAT + lane * 4;
  atomicAddF(dst + 0, hv.x * nrm);
  atomicAddF(dst + 1, hv.y * nrm);
  atomicAddF(dst + 2, hv.z * nrm);
  atomicAddF(dst + 3, hv.w * nrm);
}

// ---------------- finalize: OUT = relu(AGG + Hpre*dinv^2 + b) --------------

__global__ void gcn_finalize(const float* __restrict__ AGG,
                             const float* __restrict__ Hpre,
                             const float* __restrict__ dinv,
                             const float* __restrict__ b,
                             float* __restrict__ OUT, int n) {
  const int idx = blockIdx.x * blockDim.x + threadIdx.x;
  if (idx >= n * FEAT) return;
  const int i = idx >> 6;
  const int f = idx & (FEAT - 1);
  const float di = dinv[i];
  const float v = AGG[idx] + Hpre[idx] * (di * di) + b[f];
  OUT[idx] = v > 0.0f ? v : 0.0f;
}

// ---------------- graph pooling: G[batch[i]] += H[i] -----------------------

__global__ void gcn_pool(const float* __restrict__ H,
                         const int* __restrict__ batch,
                         float* __restrict__ G, int n) {
  const int idx = blockIdx.x * blockDim.x + threadIdx.x;
  if (idx >= n * FEAT) return;
  const int i = idx >> 6;
  const int f = idx & (FEAT - 1);
  atomicAddF(&G[(size_t)batch[i] * FEAT + f], H[idx]);
}

// ---------------- head: out[g] = dot(G[g], Wl) + bl ------------------------

__global__ void gcn_head(const float* __restrict__ G,
                         const float* __restrict__ Wl,
                         const float* __restrict__ bl,
                         float* __restrict__ out, int NG) {
  const int g = blockIdx.x * blockDim.x + threadIdx.x;
  if (g >= NG) return;
  float acc = 0.0f;
#pragma unroll
  for (int k = 0; k < FEAT; ++k) acc += G[(size_t)g * FEAT + k] * Wl[k];
  out[g] = acc + bl[0];
}

// ---------------------------------------------------------------------------

extern "C" void kernel_launch(void* const* d_in, const int* in_sizes, int n_in,
                              void* d_out, int out_size, void* d_ws, size_t ws_size,
                              hipStream_t stream) {
  const float* x    = (const float*)d_in[0];
  const int*   ei   = (const int*)d_in[1];
  const int*   bat  = (const int*)d_in[2];
  const float* W1   = (const float*)d_in[3];
  const float* b1   = (const float*)d_in[4];
  const float* W2   = (const float*)d_in[5];
  const float* b2   = (const float*)d_in[6];
  const float* Wl   = (const float*)d_in[7];
  const float* bl   = (const float*)d_in[8];
  float*       out  = (float*)d_out;

  const int n  = in_sizes[2];      // N_NODES
  const int E  = in_sizes[1] / 2;  // N_EDGES
  const int NG = out_size;         // N_GRAPHS (OUT_DIM == 1)

  // workspace carve-out (256B aligned slices)
  char* ws = (char*)d_ws;
  size_t off = 0;
  auto take = [&](size_t bytes) -> char* {
    char* p = ws + off;
    off += (bytes + 255) & ~(size_t)255;
    return p;
  };
  unsigned* deg  = (unsigned*)take((size_t)n * 4);
  float*    dinv = (float*)take((size_t)n * 4);
  float*    bufA = (float*)take((size_t)n * FEAT * 4);  // pre-activation h@W
  float*    bufB = (float*)take((size_t)n * FEAT * 4);  // agg / h1
  float*    bufC = (float*)take((size_t)n * FEAT * 4);  // agg / h2
  float*    G    = (float*)take((size_t)NG * FEAT * 4);
  (void)ws_size; (void)n_in;

  const int B   = 256;
  const int nEl = n * FEAT;
  const int gN   = (n + B - 1) / B;
  const int gE   = (E + B - 1) / B;
  const int gEl  = (nEl + B - 1) / B;
  const int gSc  = (E * 16 + B - 1) / B;     // 16 lanes per edge
  const int gMM  = (n + 127) / 128;          // 128 rows per block

  // normalization
  gcn_init_deg<<<gN, B, 0, stream>>>(deg, n);
  gcn_count_deg<<<gE, B, 0, stream>>>(ei, deg, E);
  gcn_dinv<<<gN, B, 0, stream>>>(deg, dinv, n);

  // ---- layer 1: h1 = relu(Agg(x@W1) + b1) ----
  gcn_gemm64_wmma<<<gMM, 256, 0, stream>>>(x, W1, bufA, n);
  gcn_zero_f32<<<gEl, B, 0, stream>>>(bufB, nEl);
  gcn_scatter<<<gSc, B, 0, stream>>>(ei, dinv, bufA, bufB, E);
  gcn_finalize<<<gEl, B, 0, stream>>>(bufB, bufA, dinv, b1, bufB, n);

  // ---- layer 2: h2 = relu(Agg(h1@W2) + b2) ----
  gcn_gemm64_wmma<<<gMM, 256, 0, stream>>>(bufB, W2, bufA, n);
  gcn_zero_f32<<<gEl, B, 0, stream>>>(bufC, nEl);
  gcn_scatter<<<gSc, B, 0, stream>>>(ei, dinv, bufA, bufC, E);
  gcn_finalize<<<gEl, B, 0, stream>>>(bufC, bufA, dinv, b2, bufC, n);

  // ---- pool + head ----
  gcn_zero_f32<<<(NG * FEAT + B - 1) / B, B, 0, stream>>>(G, NG * FEAT);
  gcn_pool<<<gEl, B, 0, stream>>>(bufC, bat, G, n);
  gcn_head<<<(NG + 63) / 64, 64, 0, stream>>>(G, Wl, bl, out, NG);
}